// MoELayerWrapper_84464826843919
// MI455X (gfx1250) — compile-verified
//
#include <hip/hip_runtime.h>
#include <math.h>

// ---------------- problem constants (match reference) ----------------
#define TOKENS   16384      // B*S = 4*4096
#define DIM      4096       // D
#define NEXP     64         // E
#define EPSF     1e-10f

// ---------------- GEMM tiling ----------------
#define KCHUNK          32
#define NCHUNKS         (DIM / KCHUNK)                 // 128
#define ROWS_PER_WAVE   16
#define WAVES_PER_BLK   4
#define THREADS_GEMM    (WAVES_PER_BLK * 32)           // 128
#define ROWS_PER_BLK    (WAVES_PER_BLK * ROWS_PER_WAVE)// 64
#define NBLOCKS_GEMM    (TOKENS / ROWS_PER_BLK)        // 256

// LDS strides (floats).
// As: 32 cols + 4 pad -> 16B-aligned rows, conflict-free A-fragment reads.
// Wp: K-pair interleaved W layout: row p holds {W[2p][e], W[2p+1][e]} pairs.
//     Stride 160 = 128 data + 32 pad -> consecutive pair-rows are 32 banks
//     apart, so the two half-waves (pair p vs p+1) hit disjoint bank sets.
#define AS_STRIDE 36
#define WP_STRIDE 160
#define WP_ROWS   (KCHUNK / 2)   // 16

typedef __attribute__((ext_vector_type(2))) float v2f;
typedef __attribute__((ext_vector_type(4))) float v4f;
typedef __attribute__((ext_vector_type(8))) float v8f;

// =====================================================================
// Pass 1: router GEMM (fp32 WMMA), double-buffered LDS pipeline,
// fused softmax + per-block expert-load partials.
// Each wave computes a 16x64 logits tile = 4 accumulators of 16x16.
// =====================================================================
__global__ __launch_bounds__(THREADS_GEMM)
void moe_router_gemm(const float* __restrict__ H,      // [TOKENS, DIM]
                     const float* __restrict__ W,      // [DIM, NEXP]
                     float* __restrict__ logits,       // [TOKENS, NEXP] (d_out)
                     float* __restrict__ partials)     // [NBLOCKS_GEMM, NEXP] (d_ws)
{
    __shared__ float As[2][ROWS_PER_BLK * AS_STRIDE];  // 2 x 64x32 (+pad)
    __shared__ float Wp[2][WP_ROWS * WP_STRIDE];       // 2 x 16x(128+pad)
    __shared__ float waveLoad[WAVES_PER_BLK][NEXP];

    const int tid   = threadIdx.x;
    const int wave  = tid >> 5;
    const int lane  = tid & 31;
    const int l15   = lane & 15;
    const int hi    = lane >> 4;            // 0: lanes 0-15, 1: lanes 16-31
    const int rowBlk  = blockIdx.x * ROWS_PER_BLK;
    const int waveRow = wave * ROWS_PER_WAVE;

    v8f acc[4];
#pragma unroll
    for (int t = 0; t < 4; ++t)
#pragma unroll
        for (int i = 0; i < 8; ++i) acc[t][i] = 0.0f;

    v4f aReg[4], wReg[4];

    auto loadChunk = [&](int k0) {
#pragma unroll
        for (int j = 0; j < 4; ++j) {               // A: 64 rows x 32 cols
            const int i = tid + j * THREADS_GEMM;
            const int r = i >> 3, c4 = i & 7;       // 8 float4 per row
            aReg[j] = __builtin_nontemporal_load(
                (const v4f*)(H + (size_t)(rowBlk + r) * DIM + k0 + c4 * 4));
        }
#pragma unroll
        for (int j = 0; j < 4; ++j) {               // W: 32 rows x 64 cols
            const int i = tid + j * THREADS_GEMM;
            const int r = i >> 4, c4 = i & 15;      // 16 float4 per row
            wReg[j] = *(const v4f*)(W + (size_t)(k0 + r) * NEXP + c4 * 4);
        }
    };

    auto storeChunk = [&](int buf) {
#pragma unroll
        for (int j = 0; j < 4; ++j) {
            const int i = tid + j * THREADS_GEMM;
            const int r = i >> 3, c4 = i & 7;
            *(v4f*)&As[buf][r * AS_STRIDE + c4 * 4] = aReg[j];
        }
#pragma unroll
        for (int j = 0; j < 4; ++j) {
            const int i = tid + j * THREADS_GEMM;
            const int r = i >> 4, c4 = i & 15;
            const int p = r >> 1, par = r & 1;      // pair-interleave scatter
            float* dst = &Wp[buf][p * WP_STRIDE + c4 * 8 + par];
            dst[0] = wReg[j].x; dst[2] = wReg[j].y;
            dst[4] = wReg[j].z; dst[6] = wReg[j].w;
        }
    };

    loadChunk(0);
    storeChunk(0);
    __syncthreads();

    for (int c = 0; c < NCHUNKS; ++c) {
        const int buf = c & 1;
        if (c + 1 < NCHUNKS) loadChunk((c + 1) * KCHUNK);   // prefetch next chunk

#pragma unroll
        for (int kk = 0; kk < KCHUNK; kk += 4) {
            // A frag: lanes 0-15 K=kk,kk+1; lanes 16-31 K=kk+2,kk+3
            const v2f a = *(const v2f*)&As[buf][(waveRow + l15) * AS_STRIDE + kk + hi * 2];
            const float* wrow = &Wp[buf][((kk >> 1) + hi) * WP_STRIDE + 2 * l15];
#pragma unroll
            for (int t = 0; t < 4; ++t) {
                const v2f b = *(const v2f*)(wrow + t * 32);  // contiguous b64 load
                acc[t] = __builtin_amdgcn_wmma_f32_16x16x4_f32(
                    false, a, false, b, (short)0, acc[t], false, false);
            }
        }

        __syncthreads();                       // all waves done reading buf^1
        if (c + 1 < NCHUNKS) storeChunk(buf ^ 1);
        __syncthreads();                       // stores visible before next compute
    }

    // ---- epilogue: write logits, compute softmax, accumulate expert load ----
    // C layout: VGPR r holds row (hi?8:0)+r for this half; lane covers expert t*16+l15.
    const int rowBase = rowBlk + waveRow + hi * 8;
    float loadPart[4] = {0.f, 0.f, 0.f, 0.f};

#pragma unroll
    for (int r = 0; r < 8; ++r) {
        const float v0 = acc[0][r], v1 = acc[1][r], v2 = acc[2][r], v3 = acc[3][r];
        float* out = logits + (size_t)(rowBase + r) * NEXP;
        out[     l15] = v0;
        out[16 + l15] = v1;
        out[32 + l15] = v2;
        out[48 + l15] = v3;

        // row max over 64 experts (16 lanes in this half x 4 tiles)
        float m = fmaxf(fmaxf(v0, v1), fmaxf(v2, v3));
#pragma unroll
        for (int s = 8; s >= 1; s >>= 1) m = fmaxf(m, __shfl_xor(m, s, 32));
        const float e0 = __expf(v0 - m), e1 = __expf(v1 - m);
        const float e2 = __expf(v2 - m), e3 = __expf(v3 - m);
        float ssum = e0 + e1 + e2 + e3;
#pragma unroll
        for (int s = 8; s >= 1; s >>= 1) ssum += __shfl_xor(ssum, s, 32);
        const float inv = 1.0f / ssum;
        loadPart[0] += e0 * inv;
        loadPart[1] += e1 * inv;
        loadPart[2] += e2 * inv;
        loadPart[3] += e3 * inv;
    }
    // merge the two halves (same experts, different 8 rows)
#pragma unroll
    for (int t = 0; t < 4; ++t) loadPart[t] += __shfl_xor(loadPart[t], 16, 32);
    if (hi == 0) {
#pragma unroll
        for (int t = 0; t < 4; ++t) waveLoad[wave][t * 16 + l15] = loadPart[t];
    }
    __syncthreads();
    // deterministic fixed-order per-block reduction across waves
    if (tid < NEXP) {
        float s = 0.f;
#pragma unroll
        for (int wv = 0; wv < WAVES_PER_BLK; ++wv) s += waveLoad[wv][tid];
        partials[blockIdx.x * NEXP + tid] = s;
    }
}

// =====================================================================
// Pass 2: reduce partial loads -> log(load_penalty + eps) per expert.
// =====================================================================
__global__ __launch_bounds__(NEXP)
void moe_router_penalty(const float* __restrict__ partials,  // [NBLOCKS_GEMM, NEXP]
                        float* __restrict__ logpen)          // [NEXP]
{
    __shared__ float loadsh[NEXP];
    const int e = threadIdx.x;
    float s = 0.f;
    for (int b = 0; b < NBLOCKS_GEMM; ++b) s += partials[b * NEXP + e];
    loadsh[e] = s;
    __syncthreads();
    float tot = 0.f;
#pragma unroll
    for (int i = 0; i < NEXP; ++i) tot += loadsh[i];   // fixed order, deterministic
    const float mean = tot / (float)NEXP;
    const float pen  = s / (mean + EPSF);
    logpen[e] = logf(pen + EPSF);
}

// =====================================================================
// Pass 3: adjusted = logits - logpen (in place) + top-2 per row.
// One wave per token row: lane covers experts {2*lane, 2*lane+1}.
// =====================================================================
__device__ __forceinline__ bool topk_better(float va, int ia, float vb, int ib) {
    return (va > vb) || (va == vb && ia < ib);   // jax tie-break: lower index
}

__global__ __launch_bounds__(256)
void moe_router_topk(float* __restrict__ logits,        // in/out adjusted logits
                     const float* __restrict__ logpen,  // [NEXP]
                     float* __restrict__ idx_out)       // [TOKENS, 2] (as float)
{
    const int tid  = threadIdx.x;
    const int wave = tid >> 5;
    const int lane = tid & 31;
    const int row  = blockIdx.x * 8 + wave;

    float* rowp = logits + (size_t)row * NEXP + lane * 2;
    v2f v = *(v2f*)rowp;
    const float a0 = v.x - logpen[lane * 2];
    const float a1 = v.y - logpen[lane * 2 + 1];
    v.x = a0; v.y = a1;
    *(v2f*)rowp = v;

    float b1, b2; int i1, i2;
    if (a0 >= a1) { b1 = a0; i1 = lane * 2;     b2 = a1; i2 = lane * 2 + 1; }
    else          { b1 = a1; i1 = lane * 2 + 1; b2 = a0; i2 = lane * 2;     }

#pragma unroll
    for (int s = 16; s >= 1; s >>= 1) {
        const float o1 = __shfl_xor(b1, s, 32);
        const int  oi1 = __shfl_xor(i1, s, 32);
        const float o2 = __shfl_xor(b2, s, 32);
        const int  oi2 = __shfl_xor(i2, s, 32);
        float n1, n2; int n1i, n2i;
        if (topk_better(o1, oi1, b1, i1)) {
            n1 = o1; n1i = oi1;
            if (topk_better(b1, i1, o2, oi2)) { n2 = b1; n2i = i1; }
            else                              { n2 = o2; n2i = oi2; }
        } else {
            n1 = b1; n1i = i1;
            if (topk_better(o1, oi1, b2, i2)) { n2 = o1; n2i = oi1; }
            else                              { n2 = b2; n2i = i2; }
        }
        b1 = n1; i1 = n1i; b2 = n2; i2 = n2i;
    }
    if (lane == 0) {
        idx_out[(size_t)row * 2]     = (float)i1;
        idx_out[(size_t)row * 2 + 1] = (float)i2;
    }
}

// =====================================================================
extern "C" void kernel_launch(void* const* d_in, const int* in_sizes, int n_in,
                              void* d_out, int out_size, void* d_ws, size_t ws_size,
                              hipStream_t stream) {
    (void)in_sizes; (void)n_in; (void)out_size; (void)ws_size;
    const float* H = (const float*)d_in[0];   // hidden_states [4,4096,4096] fp32
    const float* W = (const float*)d_in[1];   // w_router [4096,64] fp32

    float* logits  = (float*)d_out;                       // adjusted_logits region
    float* idx_out = logits + (size_t)TOKENS * NEXP;      // expert indices region

    float* partials = (float*)d_ws;                       // [NBLOCKS_GEMM, NEXP]
    float* logpen   = partials + NBLOCKS_GEMM * NEXP;     // [NEXP]

    moe_router_gemm<<<NBLOCKS_GEMM, THREADS_GEMM, 0, stream>>>(H, W, logits, partials);
    moe_router_penalty<<<1, NEXP, 0, stream>>>(partials, logpen);
    moe_router_topk<<<TOKENS / 8, 256, 0, stream>>>(logits, logpen, idx_out);
}